// Constant2RVLinearlayer_841813590059
// MI455X (gfx1250) — compile-verified
//
#include <hip/hip_runtime.h>
#include <cmath>

typedef __attribute__((ext_vector_type(2))) float v2f;
typedef __attribute__((ext_vector_type(4))) float v4f;
typedef __attribute__((ext_vector_type(8))) float v8f;

#define SIZE 128
#define WAVES_PER_BLOCK 8
#define ROWS_PER_WAVE 32                                   // two 16-row M tiles per wave
#define ROWS_PER_BLOCK (ROWS_PER_WAVE * WAVES_PER_BLOCK)   // 256
// pair-row stride: 144 pairs = 1152 B ≡ 128 (mod 256) -> khalf=0/1 halves hit disjoint banks
#define PAD_PAIRS 144

__device__ __forceinline__ float softplusf(float v) {
    // numerically stable log(1 + exp(v))
    return fmaxf(v, 0.0f) + log1pf(expf(-fabsf(v)));
}

// ---------------------------------------------------------------------------
// Fused kernel: mu_out = x @ W + b_mu ; sigma_out = (x^2 . sp_w) + sp_b
// Wave32 owns 32 rows x 128 cols; fp32 WMMA 16x16x4; B fragments are single
// ds_load_b64 from pair-packed LDS, each reused by both M tiles.
// ---------------------------------------------------------------------------
__global__ void __launch_bounds__(256)
rvlinear_main(const float* __restrict__ x,
              const float* __restrict__ w_mu,
              const float* __restrict__ w_sigma,
              const float* __restrict__ b_mu,
              const float* __restrict__ b_sigma,
              float* __restrict__ mu_out,
              float* __restrict__ sigma_out)
{
    __shared__ v2f   sB[(SIZE / 2) * PAD_PAIRS];   // 64 pair-rows * 144 * 8B = 73.7 KB
    __shared__ float sSpW[SIZE];
    __shared__ float sSpB[SIZE];
    __shared__ float sBMu[SIZE];

    const int tid  = threadIdx.x;
    const int lane = tid & 31;
    const int wave = tid >> 5;

    // Stage W into WMMA-B pair layout: sB[(k/2)*PAD + col] = {W[k][col], W[k+1][col]}
    for (int idx = tid; idx < (SIZE / 2) * SIZE; idx += 256) {
        const int kr2 = idx >> 7;      // k/2
        const int col = idx & 127;
        v2f p;
        p.x = w_mu[(kr2 * 2    ) * SIZE + col];
        p.y = w_mu[(kr2 * 2 + 1) * SIZE + col];
        sB[kr2 * PAD_PAIRS + col] = p;
    }
    if (tid < SIZE) {
        sSpW[tid] = softplusf(w_sigma[tid]);
        sSpB[tid] = softplusf(b_sigma[tid]);
        sBMu[tid] = b_mu[tid];
    }
    __syncthreads();

    const size_t rowbase = (size_t)blockIdx.x * ROWS_PER_BLOCK
                         + (size_t)wave * ROWS_PER_WAVE;
    const int mrow  = lane & 15;   // A row within tile / B,C column within tile
    const int khalf = lane >> 4;   // 0: K pair {k,k+1}; 1: K pair {k+2,k+3}
    const float* xrow0 = x + (rowbase + (size_t)mrow) * SIZE;
    const float* xrow1 = xrow0 + 16 * SIZE;

    // pull the streaming rows toward the WGP before the K loop (global_prefetch_b8)
    __builtin_prefetch(xrow0, 0, 1);
    __builtin_prefetch(xrow0 + 64, 0, 1);
    __builtin_prefetch(xrow1, 0, 1);
    __builtin_prefetch(xrow1 + 64, 0, 1);

    v8f  acc0[8] = {}, acc1[8] = {};
    float q0 = 0.0f, q1 = 0.0f;

    for (int k = 0; k < SIZE; k += 4) {
        const int c0 = k + 2 * khalf;
        // A fragments (ISA 16x4 f32 layout): lanes 0-15 K={k,k+1}, lanes 16-31 K={k+2,k+3}
        v2f a0 = *reinterpret_cast<const v2f*>(xrow0 + c0);
        v2f a1 = *reinterpret_cast<const v2f*>(xrow1 + c0);
        // variance path rides along on the same registers
        v2f spw = *reinterpret_cast<const v2f*>(&sSpW[c0]);
        q0 = __builtin_fmaf(a0.x * a0.x, spw.x, q0);
        q0 = __builtin_fmaf(a0.y * a0.y, spw.y, q0);
        q1 = __builtin_fmaf(a1.x * a1.x, spw.x, q1);
        q1 = __builtin_fmaf(a1.y * a1.y, spw.y, q1);

        const v2f* bb = &sB[((k >> 1) + khalf) * PAD_PAIRS + mrow];
#pragma unroll
        for (int n = 0; n < 8; ++n) {
            v2f b = bb[n * 16];   // one aligned ds_load_b64, reused by both M tiles
            acc0[n] = __builtin_amdgcn_wmma_f32_16x16x4_f32(
                          false, a0, false, b, (short)0, acc0[n], false, false);
            acc1[n] = __builtin_amdgcn_wmma_f32_16x16x4_f32(
                          false, a1, false, b, (short)0, acc1[n], false, false);
        }
    }

    // ---- mu stores: C layout: VGPR r -> M=r (lanes 0-15) / M=r+8 (lanes 16-31)
#pragma unroll
    for (int n = 0; n < 8; ++n) {
        const int col  = n * 16 + mrow;
        const float bm = sBMu[col];
#pragma unroll
        for (int r = 0; r < 8; ++r) {
            const int row = r + 8 * khalf;
            mu_out[(rowbase + row     ) * SIZE + col] = acc0[n][r] + bm;
            mu_out[(rowbase + 16 + row) * SIZE + col] = acc1[n][r] + bm;
        }
    }

    // ---- quad: lanes L / L+16 hold the two K-halves of row L -> SWAPX16 combine
    q0 += __int_as_float(__builtin_amdgcn_ds_swizzle(__float_as_int(q0), 0x401F));
    q1 += __int_as_float(__builtin_amdgcn_ds_swizzle(__float_as_int(q1), 0x401F));

    // ---- sigma: broadcast row value from lane r, one 16B store per lane per row
    v4f spb = *reinterpret_cast<const v4f*>(&sSpB[lane * 4]);
#pragma unroll
    for (int r = 0; r < 16; ++r) {
        float qa = __int_as_float(__builtin_amdgcn_ds_bpermute(r << 2, __float_as_int(q0)));
        float qb = __int_as_float(__builtin_amdgcn_ds_bpermute(r << 2, __float_as_int(q1)));
        v4f va; va.x = qa + spb.x; va.y = qa + spb.y; va.z = qa + spb.z; va.w = qa + spb.w;
        v4f vb; vb.x = qb + spb.x; vb.y = qb + spb.y; vb.z = qb + spb.z; vb.w = qb + spb.w;
        *reinterpret_cast<v4f*>(sigma_out + (rowbase + r     ) * SIZE + lane * 4) = va;
        *reinterpret_cast<v4f*>(sigma_out + (rowbase + 16 + r) * SIZE + lane * 4) = vb;
    }
}

// ---------------------------------------------------------------------------
// KL scalar:  kl = -0.5 * ( sum(log sp_w) - sum|w_mu| - sum(sp_w) )
// ---------------------------------------------------------------------------
__global__ void __launch_bounds__(256)
rvlinear_kl(const float* __restrict__ w_mu,
            const float* __restrict__ w_sigma,
            float* __restrict__ kl_out)
{
    __shared__ float red[256];
    const int tid = threadIdx.x;
    float s = 0.0f;
    for (int i = tid; i < SIZE * SIZE; i += 256) s -= fabsf(w_mu[i]);
    if (tid < SIZE) {
        const float spw = softplusf(w_sigma[tid]);
        s += logf(spw) - spw;
    }
    red[tid] = s;
    __syncthreads();
    for (int off = 128; off > 0; off >>= 1) {
        if (tid < off) red[tid] += red[tid + off];
        __syncthreads();
    }
    if (tid == 0) *kl_out = -0.5f * red[0];
}

// ---------------------------------------------------------------------------
extern "C" void kernel_launch(void* const* d_in, const int* in_sizes, int n_in,
                              void* d_out, int out_size, void* d_ws, size_t ws_size,
                              hipStream_t stream)
{
    const float* x       = (const float*)d_in[0];
    const float* w_mu    = (const float*)d_in[1];
    const float* w_sigma = (const float*)d_in[2];
    const float* b_mu    = (const float*)d_in[3];
    const float* b_sigma = (const float*)d_in[4];

    const int batch = in_sizes[0] / SIZE;

    float* out       = (float*)d_out;
    float* mu_out    = out;                               // B*128
    float* sigma_out = out + (size_t)batch * SIZE;        // B*128
    float* kl_out    = out + (size_t)batch * SIZE * 2;    // 1

    rvlinear_kl<<<1, 256, 0, stream>>>(w_mu, w_sigma, kl_out);
    rvlinear_main<<<batch / ROWS_PER_BLOCK, 256, 0, stream>>>(
        x, w_mu, w_sigma, b_mu, b_sigma, mu_out, sigma_out);
}